// GAT_21569325760839
// MI455X (gfx1250) — compile-verified
//
#include <hip/hip_runtime.h>

// ---------------------------------------------------------------------------
// Two-layer GAT for MI455X (gfx1250, wave32).
//  - Dense projections: V_WMMA_F32_16X16X4_F32, B operand staged in LDS by the
//    Tensor Data Mover (tensor_load_to_lds + s_wait_tensorcnt), bank-conflict-
//    free padded layout (TDM hardware padding), register-batched fragments.
//  - Edge softmax + aggregation: global float atomics (L2-resident).
// ---------------------------------------------------------------------------

namespace {
constexpr int kN    = 50000;            // nodes
constexpr int kE    = 800000;           // raw edges
constexpr int kEtot = kE + kN;          // + self loops (PyG semantics)
constexpr int kF    = 256;              // input features
constexpr int kH1   = 8;                // heads, layer 1
constexpr int kC1   = 32;               // channels/head, layer 1
constexpr int kHid  = kH1 * kC1;        // 256
constexpr int kH2   = 1;                // heads, layer 2
constexpr int kC2   = 128;              // channels, layer 2
constexpr float kSlope = 0.2f;
}

typedef __attribute__((ext_vector_type(2))) float        v2f;
typedef __attribute__((ext_vector_type(8))) float        v8f;
typedef __attribute__((ext_vector_type(4))) unsigned int v4u;
typedef __attribute__((ext_vector_type(8))) int          v8i;
typedef __attribute__((ext_vector_type(4))) int          v4i;

__device__ __forceinline__ float lrelu(float x) { return x >= 0.0f ? x : kSlope * x; }

// float atomic max on one cell: signed-max for non-negative payloads,
// unsigned-min for negative ones (init cell to -inf = 0xFF800000).
__device__ __forceinline__ void atomic_fmax(float* addr, float v) {
    if (v >= 0.0f) atomicMax((int*)addr, __float_as_int(v));
    else           atomicMin((unsigned int*)addr, __float_as_uint(v));
}

// ---------------------------------------------------------------------------
// WMMA fp32 GEMM: Out[M,OC] = A[M,K] * W[OC,K]^T.
// W is DMA'd into LDS once per block by the TDM with a padded row stride of
// K+4 floats (pad_interval=256 DWORDs, pad_amount=4 DWORDs -> requires K==256,
// which holds for both layers). Row stride K+4 makes the half-wave ds_load_b64
// pattern cover all 64 LDS banks exactly once (4*c + kk spacing).
// One wave computes a 16x64 strip (4 accumulators); fragments are preloaded in
// register batches of 8 k-steps so waits amortize over 32-WMMA chains.
// Requires M%16==0, K==256, OC%64==0 (holds: 50000, 256, {256,128}).
// ---------------------------------------------------------------------------
__global__ void gat_gemm_wmma(const float* __restrict__ A, const float* __restrict__ W,
                              float* __restrict__ Out, int M, int K, int OC) {
    extern __shared__ float ldsW[];     // OC * (K+4) floats

    // ---- Stage W -> LDS via the Tensor Data Mover (wave 0 issues the DMA) ---
    if (threadIdx.x < 32) {
        const unsigned ldsOff = (unsigned)(unsigned long long)(void*)ldsW;
        const unsigned long long ga = (unsigned long long)(const void*)W;

        // D# group 0: count=1 | lds_addr | global_addr[56:0] | type=2
        v4u g0;
        g0.x = 1u;                                           // count=1, user mode
        g0.y = ldsOff;                                       // LDS byte address
        g0.z = (unsigned)ga;                                 // global addr [31:0]
        g0.w = (unsigned)((ga >> 32) & 0x1FFFFFFull) | (2u << 30);  // [56:32] | type=2

        // D# group 1: data_size=4B, pad_enable, pad_interval=256dw, pad_amount=4dw,
        // tensor_dim0=K, tensor_dim1=OC, tile_dim0=K, tile_dim1=OC, dim0_stride=K.
        v8i g1;
        g1[0] = (int)((2u << 16) | (1u << 20) | (7u << 22) | (3u << 25));
        g1[1] = (int)((unsigned)K << 16);    // [63:48] = tensor_dim0 lo16
        g1[2] = (int)((unsigned)OC << 16);   // [79:64]=dim0 hi16(0), [95:80]=tensor_dim1 lo16
        g1[3] = (int)((unsigned)K << 16);    // [111:96]=dim1 hi16(0), [127:112]=tile_dim0
        g1[4] = OC;                          // [143:128]=tile_dim1, [159:144]=tile_dim2(0)
        g1[5] = K;                           // [191:160]=tensor_dim0_stride lo32
        g1[6] = 0;                           // stride hi16, dim1_stride lo16
        g1[7] = 0;                           // dim1_stride hi32
        v4i g2 = {0, 0, 0, 0};               // dims 2..4 unused (2D tensor)
        v4i g3 = {0, 0, 0, 0};
        v8i g4 = {0, 0, 0, 0, 0, 0, 0, 0};   // extra group in 6-arg toolchain form

        __builtin_amdgcn_tensor_load_to_lds(g0, g1, g2, g3, g4, 0);
        __builtin_amdgcn_s_wait_tensorcnt(0);
    }
    __syncthreads();

    // ---- WMMA tiles ---------------------------------------------------------
    const int strips     = OC >> 6;                 // 64-wide column strips
    const int rowTiles   = M >> 4;
    const int totalTiles = rowTiles * strips;
    const int wavesTotal = (gridDim.x * blockDim.x) >> 5;
    const int waveG      = (blockIdx.x * blockDim.x + threadIdx.x) >> 5;
    const int lane = threadIdx.x & 31;
    const int half = lane >> 4;                     // 0: lanes 0-15, 1: lanes 16-31
    const int l15  = lane & 15;
    const int ldsStride = K + 4;

    for (int tile = waveG; tile < totalTiles; tile += wavesTotal) {  // wave-uniform
        const int rowTile = tile / strips;
        const int strip   = tile % strips;

        const float* aRow = A + (size_t)(rowTile * 16 + l15) * K;
        const float* b0p  = ldsW + (size_t)(strip * 64 +  0 + l15) * ldsStride;
        const float* b1p  = b0p + (size_t)16 * ldsStride;
        const float* b2p  = b0p + (size_t)32 * ldsStride;
        const float* b3p  = b0p + (size_t)48 * ldsStride;

        v8f acc0 = {}, acc1 = {}, acc2 = {}, acc3 = {};
        for (int kb = 0; kb < K; kb += 32) {
            v2f a[8], bb0[8], bb1[8], bb2[8], bb3[8];
#pragma unroll
            for (int i = 0; i < 8; ++i)             // batched global loads (A)
                a[i] = *(const v2f*)(aRow + kb + i * 4 + half * 2);
#pragma unroll
            for (int i = 0; i < 8; ++i) {           // batched LDS loads (B)
                const int kk = kb + i * 4 + half * 2;
                bb0[i] = *(const v2f*)(b0p + kk);
                bb1[i] = *(const v2f*)(b1p + kk);
                bb2[i] = *(const v2f*)(b2p + kk);
                bb3[i] = *(const v2f*)(b3p + kk);
            }
#pragma unroll
            for (int i = 0; i < 8; ++i) {           // 32-WMMA chain per batch
                acc0 = __builtin_amdgcn_wmma_f32_16x16x4_f32(false, a[i], false, bb0[i], (short)0, acc0, false, false);
                acc1 = __builtin_amdgcn_wmma_f32_16x16x4_f32(false, a[i], false, bb1[i], (short)0, acc1, false, false);
                acc2 = __builtin_amdgcn_wmma_f32_16x16x4_f32(false, a[i], false, bb2[i], (short)0, acc2, false, false);
                acc3 = __builtin_amdgcn_wmma_f32_16x16x4_f32(false, a[i], false, bb3[i], (short)0, acc3, false, false);
            }
        }

        const int rowBase = rowTile * 16 + half * 8;
        float* o = Out + (size_t)rowBase * OC + strip * 64 + l15;
#pragma unroll
        for (int r = 0; r < 8; ++r) {
            o[(size_t)r * OC +  0] = acc0[r];
            o[(size_t)r * OC + 16] = acc1[r];
            o[(size_t)r * OC + 32] = acc2[r];
            o[(size_t)r * OC + 48] = acc3[r];
        }
    }
}

// Per-(node,head) attention scalars: alS = <h, a_src>, alD = <h, a_dst>.
__global__ void gat_node_attn(const float* __restrict__ h, const float* __restrict__ aS,
                              const float* __restrict__ aD, float* __restrict__ alS,
                              float* __restrict__ alD, int H, int C) {
    const int t = blockIdx.x * blockDim.x + threadIdx.x;
    if (t >= kN * H) return;
    const int hh = t % H;
    const float* hp = h + (size_t)t * C;   // t = n*H+hh; [n][h][c] row-major
    const float* sp = aS + hh * C;
    const float* dp = aD + hh * C;
    float s = 0.0f, d = 0.0f;
    for (int c = 0; c < C; ++c) { const float v = hp[c]; s = fmaf(v, sp[c], s); d = fmaf(v, dp[c], d); }
    alS[t] = s;
    alD[t] = d;
}

// acc[n,j] = bias[j]; emax = -inf; den = 0.
__global__ void gat_init(float* __restrict__ acc, const float* __restrict__ bias,
                         float* __restrict__ emax, float* __restrict__ den, int HC, int H) {
    const long long t = (long long)blockIdx.x * blockDim.x + threadIdx.x;
    if (t >= (long long)kN * HC) return;
    acc[t] = bias[t % HC];
    if (t < (long long)kN * H) { emax[t] = __int_as_float(0xFF800000); den[t] = 0.0f; }
}

__device__ __forceinline__ void edge_sd(const int* __restrict__ src, const int* __restrict__ dst,
                                        int e, int& s, int& d) {
    if (e < kE) { s = src[e]; d = dst[e]; } else { s = e - kE; d = s; }  // self loops
}

// Pass 1: segment max of LeakyReLU logits over dst.
__global__ void gat_edge_max(const int* __restrict__ src, const int* __restrict__ dst,
                             const float* __restrict__ alS, const float* __restrict__ alD,
                             float* __restrict__ emax, int H) {
    const int t = blockIdx.x * blockDim.x + threadIdx.x;
    if (t >= kEtot * H) return;
    const int e = t / H, hh = t - e * H;
    int s, d; edge_sd(src, dst, e, s, d);
    const float ev = lrelu(alS[s * H + hh] + alD[d * H + hh]);
    atomic_fmax(&emax[d * H + hh], ev);
}

// Pass 2: segment sum of exp(e - max).
__global__ void gat_edge_expsum(const int* __restrict__ src, const int* __restrict__ dst,
                                const float* __restrict__ alS, const float* __restrict__ alD,
                                const float* __restrict__ emax, float* __restrict__ den, int H) {
    const int t = blockIdx.x * blockDim.x + threadIdx.x;
    if (t >= kEtot * H) return;
    const int e = t / H, hh = t - e * H;
    int s, d; edge_sd(src, dst, e, s, d);
    const int idx = d * H + hh;
    const float ev = lrelu(alS[s * H + hh] + alD[idx]);
    atomicAdd(&den[idx], __expf(ev - emax[idx]));
}

// Pass 3: acc[dst] += alpha * h[src].  One thread per (edge, head, 32-ch chunk).
__global__ void gat_edge_msg(const int* __restrict__ src, const int* __restrict__ dst,
                             const float* __restrict__ alS, const float* __restrict__ alD,
                             const float* __restrict__ emax, const float* __restrict__ den,
                             const float* __restrict__ h, float* __restrict__ acc,
                             int H, int C, int chunks) {
    const long long t = (long long)blockIdx.x * blockDim.x + threadIdx.x;
    if (t >= (long long)kEtot * H * chunks) return;
    const int ck = (int)(t % chunks);
    const long long t2 = t / chunks;
    const int hh = (int)(t2 % H);
    const int e  = (int)(t2 / H);
    int s, d; edge_sd(src, dst, e, s, d);
    const int idx = d * H + hh;
    const float ev    = lrelu(alS[s * H + hh] + alD[idx]);
    const float alpha = __expf(ev - emax[idx]) / (den[idx] + 1e-16f);
    const float* hp = h   + ((size_t)s * H + hh) * C + ck * 32;
    float*       ap = acc + ((size_t)d * H + hh) * C + ck * 32;
#pragma unroll
    for (int c = 0; c < 32; ++c) atomicAdd(&ap[c], alpha * hp[c]);
}

__global__ void gat_lrelu(float* __restrict__ buf, int n) {
    const int t = blockIdx.x * blockDim.x + threadIdx.x;
    if (t < n) { const float v = buf[t]; buf[t] = v >= 0.0f ? v : kSlope * v; }
}

// ---------------------------------------------------------------------------
extern "C" void kernel_launch(void* const* d_in, const int* in_sizes, int n_in,
                              void* d_out, int out_size, void* d_ws, size_t ws_size,
                              hipStream_t stream) {
    const int*   src = (const int*)d_in[0];        // edge_index row 0
    const int*   dst = src + kE;                   // edge_index row 1
    const float* x   = (const float*)d_in[1];      // [N,256]
    const float* W1  = (const float*)d_in[2];      // [256,256] (out,in)
    const float* aS1 = (const float*)d_in[3];      // [8,32]
    const float* aD1 = (const float*)d_in[4];      // [8,32]
    const float* b1  = (const float*)d_in[5];      // [256]
    const float* W2  = (const float*)d_in[6];      // [128,256]
    const float* aS2 = (const float*)d_in[7];      // [1,128]
    const float* aD2 = (const float*)d_in[8];      // [1,128]
    const float* b2  = (const float*)d_in[9];      // [128]
    float* out = (float*)d_out;                    // [N,128] fp32

    // workspace layout (floats): ~135 MB total (fits in the 192 MB L2)
    float* p     = (float*)d_ws;
    float* h1    = p; p += (size_t)kN * kHid;      // 51.2 MB
    float* acc1  = p; p += (size_t)kN * kHid;      // 51.2 MB (becomes layer-2 input)
    float* als1  = p; p += (size_t)kN * kH1;
    float* ald1  = p; p += (size_t)kN * kH1;
    float* emax1 = p; p += (size_t)kN * kH1;
    float* den1  = p; p += (size_t)kN * kH1;
    float* g     = p; p += (size_t)kN * kC2;       // 25.6 MB
    float* als2  = p; p += kN;
    float* ald2  = p; p += kN;
    float* emax2 = p; p += kN;
    float* den2  = p; p += kN;
    (void)ws_size; (void)in_sizes; (void)n_in; (void)out_size;

    const int TPB = 256;
    const int GEMM_BLOCKS = 256;                   // persistent waves, W staged once/block
    auto cdiv = [](long long a, long long b) { return (int)((a + b - 1) / b); };

    // ----- Layer 1 -----
    {
        const size_t ldsBytes = (size_t)kHid * (kF + 4) * sizeof(float);   // 260 KB
        gat_gemm_wmma<<<GEMM_BLOCKS, TPB, ldsBytes, stream>>>(x, W1, h1, kN, kF, kHid);
    }
    gat_node_attn<<<cdiv((long long)kN * kH1, TPB), TPB, 0, stream>>>(h1, aS1, aD1, als1, ald1, kH1, kC1);
    gat_init<<<cdiv((long long)kN * kHid, TPB), TPB, 0, stream>>>(acc1, b1, emax1, den1, kHid, kH1);
    gat_edge_max<<<cdiv((long long)kEtot * kH1, TPB), TPB, 0, stream>>>(src, dst, als1, ald1, emax1, kH1);
    gat_edge_expsum<<<cdiv((long long)kEtot * kH1, TPB), TPB, 0, stream>>>(src, dst, als1, ald1, emax1, den1, kH1);
    gat_edge_msg<<<cdiv((long long)kEtot * kH1, TPB), TPB, 0, stream>>>(src, dst, als1, ald1, emax1, den1,
                                                                        h1, acc1, kH1, kC1, /*chunks=*/1);
    gat_lrelu<<<cdiv((long long)kN * kHid, TPB), TPB, 0, stream>>>(acc1, kN * kHid);

    // ----- Layer 2 -----
    {
        const size_t ldsBytes = (size_t)kC2 * (kHid + 4) * sizeof(float);  // 130 KB
        gat_gemm_wmma<<<GEMM_BLOCKS, TPB, ldsBytes, stream>>>(acc1, W2, g, kN, kHid, kC2);
    }
    gat_node_attn<<<cdiv((long long)kN * kH2, TPB), TPB, 0, stream>>>(g, aS2, aD2, als2, ald2, kH2, kC2);
    gat_init<<<cdiv((long long)kN * kC2, TPB), TPB, 0, stream>>>(out, b2, emax2, den2, kC2, kH2);
    gat_edge_max<<<cdiv((long long)kEtot * kH2, TPB), TPB, 0, stream>>>(src, dst, als2, ald2, emax2, kH2);
    gat_edge_expsum<<<cdiv((long long)kEtot * kH2, TPB), TPB, 0, stream>>>(src, dst, als2, ald2, emax2, den2, kH2);
    gat_edge_msg<<<cdiv((long long)kEtot * kH2 * 4, TPB), TPB, 0, stream>>>(src, dst, als2, ald2, emax2, den2,
                                                                            g, out, kH2, kC2, /*chunks=*/4);
}